// EquiForwardModel_3066606649477
// MI455X (gfx1250) — compile-verified
//
#include <hip/hip_runtime.h>
#include <hip/hip_bf16.h>

typedef float v2f __attribute__((ext_vector_type(2)));
typedef float v8f __attribute__((ext_vector_type(8)));

#define H 128
#define KDIM 128
#define CHUNK 40960
#define BSTRIDE 132   // padded LDS row stride (floats): 132%64=4 -> halves hit disjoint banks

// -------- float ordered-int encoding for atomic max --------
__device__ __forceinline__ int ford(float f) {
    int b = __float_as_int(f);
    return b >= 0 ? b : (b ^ 0x7fffffff);
}
__device__ __forceinline__ float forddec(int k) {
    return __int_as_float(k >= 0 ? k : (k ^ 0x7fffffff));
}
__device__ __forceinline__ float siluf(float x) { return x / (1.0f + expf(-x)); }

// ================= generic fp32 WMMA GEMM (K fixed at 128) =================
// C[M,N] = A[M,128] @ B[128,N]                        (mode 0)
// C = A@B + addend + rowscale[m]*colbias[n]           (mode 1, node update)
// one wave per 16x16 output tile; N multiple of 16.
__global__ __launch_bounds__(32) void gemm_wmma_f32(
    const float* __restrict__ A, int lda,
    const float* __restrict__ B, int ldb,
    float* __restrict__ C, int ldc,
    int M, int N, int mode,
    const float* __restrict__ addend,
    const float* __restrict__ rowscale,
    const float* __restrict__ colbias)
{
    int tilesN = N >> 4;
    int tm = blockIdx.x / tilesN;
    int tn = blockIdx.x % tilesN;
    int lane = threadIdx.x;
    int half = lane >> 4;      // 0: K+0/1, 1: K+2/3
    int l16  = lane & 15;
    int m_a = tm * 16 + l16;
    int mc  = m_a < M ? m_a : (M - 1);
    const float* Arow = A + (size_t)mc * lda;
    int n_b = tn * 16 + l16;
    v8f acc = {};
#pragma unroll 8
    for (int k0 = 0; k0 < KDIM; k0 += 4) {
        int ka = k0 + half * 2;
        v2f a, b;
        a.x = Arow[ka];
        a.y = Arow[ka + 1];
        b.x = B[(size_t)ka * ldb + n_b];
        b.y = B[(size_t)(ka + 1) * ldb + n_b];
        acc = __builtin_amdgcn_wmma_f32_16x16x4_f32(false, a, false, b,
                                                    (short)0, acc, false, false);
    }
    int n = tn * 16 + l16;
    int mbase = tm * 16 + half * 8;
#pragma unroll
    for (int r = 0; r < 8; ++r) {
        int m = mbase + r;
        if (m < M) {
            float v = acc[r];
            if (mode == 1)
                v += addend[(size_t)m * ldc + n] + rowscale[m] * colbias[n];
            C[(size_t)m * ldc + n] = v;
        }
    }
}

// ===== edge GEMM: T[nE,128] = relu(a_e*w0+b0) @ Bm[128,128] =====
// 256 threads = 8 waves; block handles one 16-row M tile across all 8 N tiles.
// B panel staged to LDS once per block via async global->LDS (ASYNCcnt),
// A fragment generated in registers from the scalar edge_attr.
__global__ __launch_bounds__(256) void edge_gemm_wmma(
    const float* __restrict__ edge_attr, int eStart, int nE,
    const float* __restrict__ w0, const float* __restrict__ b0,
    const float* __restrict__ Bm, float* __restrict__ Tc)
{
    __shared__ float Bs[H * BSTRIDE];
    int t = threadIdx.x;

    // ---- async stage B (128x128 f32, 64KB) into LDS, padded row stride ----
    {
        unsigned ldsbase = (unsigned)(uintptr_t)(&Bs[0]);
        unsigned long long sa = (unsigned long long)(uintptr_t)Bm;
#pragma unroll
        for (int i = 0; i < 16; ++i) {
            int idx = i * 256 + t;          // float4 chunk id, 4096 total
            int row = idx >> 5;             // 32 chunks (of 4 floats) per row
            int c4  = (idx & 31) << 2;      // starting float column
            unsigned goff = (unsigned)((row * H + c4) * 4);
            unsigned laddr = ldsbase + (unsigned)((row * BSTRIDE + c4) * 4);
            asm volatile("global_load_async_to_lds_b128 %0, %1, %2"
                         :: "v"(laddr), "v"(goff), "s"(sa)
                         : "memory");
        }
        asm volatile("s_wait_asynccnt 0" ::: "memory");
    }
    __syncthreads();

    int wave = t >> 5;          // = tn (0..7)
    int lane = t & 31;
    int half = lane >> 4;
    int l16  = lane & 15;
    int tm   = blockIdx.x;
    int mloc = tm * 16 + l16;
    float aE = (mloc < nE) ? edge_attr[eStart + mloc] : 0.0f;
    int n_b  = wave * 16 + l16;
    v8f acc = {};
#pragma unroll 8
    for (int k0 = 0; k0 < KDIM; k0 += 4) {
        int ka = k0 + half * 2;
        v2f a, b;
        a.x = fmaxf(aE * w0[ka] + b0[ka], 0.0f);
        a.y = fmaxf(aE * w0[ka + 1] + b0[ka + 1], 0.0f);
        b.x = Bs[ka * BSTRIDE + n_b];
        b.y = Bs[(ka + 1) * BSTRIDE + n_b];
        acc = __builtin_amdgcn_wmma_f32_16x16x4_f32(false, a, false, b,
                                                    (short)0, acc, false, false);
    }
    int mbase = tm * 16 + half * 8;
#pragma unroll
    for (int r = 0; r < 8; ++r)
        Tc[(size_t)(mbase + r) * H + wave * 16 + l16] = acc[r];
}

// ================= small kernels =================
__global__ void graph_emb_kernel(const float* __restrict__ gf,
                                 const float* __restrict__ Wge,
                                 const float* __restrict__ bge,
                                 float* __restrict__ gemb)
{
    int g = blockIdx.x, j = threadIdx.x;
    float acc = bge[j];
    for (int i = 0; i < 5; ++i) acc += gf[g * 5 + i] * Wge[i * H + j];
    gemb[g * H + j] = acc;
}

__global__ void node_enc_kernel(const float* __restrict__ x,
                                const float* __restrict__ Wne,
                                const float* __restrict__ bne,
                                const float* __restrict__ gemb,
                                const int* __restrict__ batch,
                                float* __restrict__ h, int NN)
{
    int idx = blockIdx.x * blockDim.x + threadIdx.x;
    if (idx >= NN * H) return;
    int n = idx >> 7, j = idx & 127;
    float acc = bne[j] + gemb[batch[n] * H + j];
    for (int i = 0; i < 5; ++i) acc += x[n * 5 + i] * Wne[i * H + j];
    h[idx] = acc;
}

// folded biases: Battn = bq@Wa0_q + bk@Wa0_k + be1@Wa0_e + ba0
//                Bmsg  = bv@Wm0_v + be1@Wm0_e + bm0
__global__ void fold_bias_kernel(const float* bq, const float* bk,
                                 const float* bv, const float* be1,
                                 const float* Wa0, const float* ba0,
                                 const float* Wm0, const float* bm0,
                                 float* Battn, float* Bmsg)
{
    int j = threadIdx.x;
    float a = ba0[j], m = bm0[j];
    for (int i = 0; i < H; ++i) {
        a += bq[i] * Wa0[i * H + j]
           + bk[i] * Wa0[(H + i) * H + j]
           + be1[i] * Wa0[(2 * H + i) * H + j];
        m += bv[i] * Wm0[i * H + j]
           + be1[i] * Wm0[(H + i) * H + j];
    }
    Battn[j] = a;
    Bmsg[j]  = m;
}

__global__ void init_accum_kernel(float* __restrict__ S, float* __restrict__ wsum,
                                  float* __restrict__ den, int* __restrict__ maxint,
                                  int NN)
{
    int idx = blockIdx.x * blockDim.x + threadIdx.x;
    if (idx < NN * H) S[idx] = 0.0f;
    if (idx < NN) { wsum[idx] = 0.0f; den[idx] = 0.0f; maxint[idx] = ford(-INFINITY); }
}

// wave per edge: s = relu(Aq[col]+Ak[row]+Ae+Battn) . Wa1 + ba1; atomicMax per col
__global__ __launch_bounds__(32) void attn_s_kernel(
    const float* __restrict__ PN, const float* __restrict__ Tc,
    const int* __restrict__ ei, int Etot, int eStart,
    const float* __restrict__ Battn, const float* __restrict__ Wa1,
    const float* __restrict__ ba1, float* __restrict__ sbuf,
    int* __restrict__ maxint)
{
    int el = blockIdx.x;
    int e = eStart + el;
    int lane = threadIdx.x;
    int row = ei[e], col = ei[Etot + e];
    int j0 = lane * 4;
    float4 aq = *(const float4*)(PN + (size_t)col * 384 + j0);
    float4 ak = *(const float4*)(PN + (size_t)row * 384 + 128 + j0);
    float4 ae = *(const float4*)(Tc + (size_t)el * H + j0);
    float4 bb = *(const float4*)(Battn + j0);
    float4 wv = *(const float4*)(Wa1 + j0);
    float p = fmaxf(aq.x + ak.x + ae.x + bb.x, 0.f) * wv.x
            + fmaxf(aq.y + ak.y + ae.y + bb.y, 0.f) * wv.y
            + fmaxf(aq.z + ak.z + ae.z + bb.z, 0.f) * wv.z
            + fmaxf(aq.w + ak.w + ae.w + bb.w, 0.f) * wv.w;
    for (int off = 16; off; off >>= 1) p += __shfl_xor(p, off, 32);
    if (lane == 0) {
        float s = p + ba1[0];
        sbuf[e] = s;
        atomicMax(maxint + col, ford(s));
    }
}

__global__ void expden_kernel(const float* __restrict__ sbuf,
                              const int* __restrict__ ei, int Etot,
                              const int* __restrict__ maxint,
                              float* __restrict__ den, float* __restrict__ exbuf)
{
    int e = blockIdx.x * blockDim.x + threadIdx.x;
    if (e >= Etot) return;
    int col = ei[Etot + e];
    float ex = expf(sbuf[e] - forddec(maxint[col]));
    exbuf[e] = ex;
    atomicAdd(den + col, ex);
}

// wave per edge: m1 = silu(Mv[row]+Me+Bmsg); S[col]+=w*m1; wsum[col]+=w
__global__ __launch_bounds__(32) void msg_scatter_kernel(
    const float* __restrict__ PN, const float* __restrict__ Tc,
    const int* __restrict__ ei, int Etot, int eStart,
    const float* __restrict__ Bmsg, const float* __restrict__ exbuf,
    const float* __restrict__ den, float* __restrict__ S,
    float* __restrict__ wsum)
{
    int el = blockIdx.x;
    int e = eStart + el;
    int lane = threadIdx.x;
    int row = ei[e], col = ei[Etot + e];
    float w = exbuf[e] / (den[col] + 1e-16f);
    int j0 = lane * 4;
    float4 mv = *(const float4*)(PN + (size_t)row * 384 + 256 + j0);
    float4 me = *(const float4*)(Tc + (size_t)el * H + j0);
    float4 bb = *(const float4*)(Bmsg + j0);
    float t0 = mv.x + me.x + bb.x, t1 = mv.y + me.y + bb.y;
    float t2 = mv.z + me.z + bb.z, t3 = mv.w + me.w + bb.w;
    float* sp = S + (size_t)col * H + j0;
    atomicAdd(sp + 0, w * siluf(t0));
    atomicAdd(sp + 1, w * siluf(t1));
    atomicAdd(sp + 2, w * siluf(t2));
    atomicAdd(sp + 3, w * siluf(t3));
    if (lane == 0) atomicAdd(wsum + col, w);
}

__global__ void zero_pool_kernel(float* hs, float* cnt, int G)
{
    int idx = blockIdx.x * blockDim.x + threadIdx.x;
    if (idx < G * H) hs[idx] = 0.0f;
    if (idx < G) cnt[idx] = 0.0f;
}

__global__ __launch_bounds__(32) void pool_scatter_kernel(
    const float* __restrict__ h, const int* __restrict__ batch,
    float* __restrict__ hs, float* __restrict__ cnt)
{
    int n = blockIdx.x, lane = threadIdx.x;
    int g = batch[n];
    int j0 = lane * 4;
    float4 v = *(const float4*)(h + (size_t)n * H + j0);
    float* p = hs + (size_t)g * H + j0;
    atomicAdd(p + 0, v.x); atomicAdd(p + 1, v.y);
    atomicAdd(p + 2, v.z); atomicAdd(p + 3, v.w);
    if (lane == 0) atomicAdd(cnt + g, 1.0f);
}

__global__ void pool_final_kernel(const float* hs, const float* cnt,
                                  float* hg, int G)
{
    int idx = blockIdx.x * blockDim.x + threadIdx.x;
    if (idx >= G * H) return;
    hg[idx] = hs[idx] / fmaxf(cnt[idx >> 7], 1.0f);
}

// ============ readout heads ============
struct HeadArgs {
    const float* W0; const float* B0;
    const float* W1; const float* B1;
    const float* W2; const float* B2;
    const float* W3; const float* B3;
    int nl; int col;
};

__global__ void head_kernel(const float* __restrict__ hg, HeadArgs a,
                            float* __restrict__ out)
{
    __shared__ float cur[H];
    int g = blockIdx.x, j = threadIdx.x;
    cur[j] = hg[g * H + j];
    __syncthreads();
    const float* Ws[4] = { a.W0, a.W1, a.W2, a.W3 };
    const float* Bs[4] = { a.B0, a.B1, a.B2, a.B3 };
    int dims[5];
    if (a.nl == 4) { dims[0]=128; dims[1]=128; dims[2]=64; dims[3]=32; dims[4]=1; }
    else           { dims[0]=128; dims[1]=64;  dims[2]=32; dims[3]=1;  dims[4]=1; }
    for (int l = 0; l < a.nl; ++l) {
        int din = dims[l], dout = dims[l + 1];
        float acc = 0.0f;
        if (j < dout) {
            acc = Bs[l][j];
            for (int i = 0; i < din; ++i) acc += cur[i] * Ws[l][i * dout + j];
            if (l < a.nl - 1) acc = siluf(acc);
        }
        __syncthreads();
        if (j < dout) cur[j] = acc;
        __syncthreads();
    }
    if (j == 0) out[g * 5 + a.col] = cur[0];
}

// ================= host side =================
extern "C" void kernel_launch(void* const* d_in, const int* in_sizes, int n_in,
                              void* d_out, int out_size, void* d_ws, size_t ws_size,
                              hipStream_t stream)
{
    const float* x     = (const float*)d_in[0];
    const float* gf    = (const float*)d_in[2];
    const int*   batch = (const int*)d_in[3];
    const int*   ei    = (const int*)d_in[4];
    const float* ea    = (const float*)d_in[5];
    auto P = [&](int i) { return (const float*)d_in[6 + i]; };

    const int NN = in_sizes[0] / 5;
    const int E  = in_sizes[5];
    const int G  = in_sizes[2] / 5;

    // carve workspace (256B aligned regions)
    char* wsb = (char*)d_ws;
    size_t off = 0;
    auto carve = [&](size_t bytes) {
        off = (off + 255) & ~(size_t)255;
        size_t o = off; off += bytes; return (void*)(wsb + o);
    };
    float* hA    = (float*)carve((size_t)NN * H * 4);
    float* hB    = (float*)carve((size_t)NN * H * 4);
    float* PNbuf = (float*)carve((size_t)NN * 384 * 4);
    float* Sbuf  = (float*)carve((size_t)NN * H * 4);
    float* Tc    = (float*)carve((size_t)CHUNK * H * 4);
    float* sbuf  = (float*)carve((size_t)E * 4);
    float* exbuf = (float*)carve((size_t)E * 4);
    int*   maxint= (int*)  carve((size_t)NN * 4);
    float* den   = (float*)carve((size_t)NN * 4);
    float* wsum  = (float*)carve((size_t)NN * 4);
    float* W3    = (float*)carve((size_t)H * 384 * 4);
    float* Ca    = (float*)carve((size_t)H * H * 4);
    float* Cm    = (float*)carve((size_t)H * H * 4);
    float* Battn = (float*)carve(H * 4);
    float* Bmsg  = (float*)carve(H * 4);
    float* gemb  = (float*)carve((size_t)G * H * 4);
    float* hs    = (float*)carve((size_t)G * H * 4);
    float* cnt   = (float*)carve((size_t)G * 4);
    float* hg    = (float*)carve((size_t)G * H * 4);
    (void)ws_size; (void)n_in; (void)out_size;

    // node encoding:  h = x@Wne + bne + (gf@Wge + bge)[batch]
    graph_emb_kernel<<<G, H, 0, stream>>>(gf, P(90), P(91), gemb);
    node_enc_kernel<<<(NN * H + 255) / 256, 256, 0, stream>>>(
        x, P(126), P(127), gemb, batch, hA, NN);

    const int tilesM_nodes = (NN + 15) / 16;
    float* hin = hA; float* hout = hB;

    for (int b = 0; b < 5; ++b) {
        int pb = b * 18;
        const float* Wa0 = P(pb + 0);   // [384,128]
        const float* ba0 = P(pb + 1);
        const float* Wa1 = P(pb + 2);   // [128,1]
        const float* ba1 = P(pb + 3);
        const float* w0  = P(pb + 4);   // edge0.W [1,128]
        const float* b0  = P(pb + 5);
        const float* We1 = P(pb + 6);   // edge1.W [128,128]
        const float* be1 = P(pb + 7);
        const float* Wk  = P(pb + 8);
        const float* bk  = P(pb + 9);
        const float* Wm0 = P(pb + 10);  // [256,128]
        const float* bm0 = P(pb + 11);
        const float* Wm1 = P(pb + 12);  // [128,128]
        const float* bm1 = P(pb + 13);
        const float* Wq  = P(pb + 14);
        const float* bq  = P(pb + 15);
        const float* Wv  = P(pb + 16);
        const float* bv  = P(pb + 17);

        // --- fold weights (128x128x128 WMMA GEMMs) ---
        gemm_wmma_f32<<<64, 32, 0, stream>>>(Wq, H, Wa0,           H, W3,       384, H, H, 0, nullptr, nullptr, nullptr);
        gemm_wmma_f32<<<64, 32, 0, stream>>>(Wk, H, Wa0 + H * H,   H, W3 + 128, 384, H, H, 0, nullptr, nullptr, nullptr);
        gemm_wmma_f32<<<64, 32, 0, stream>>>(Wv, H, Wm0,           H, W3 + 256, 384, H, H, 0, nullptr, nullptr, nullptr);
        gemm_wmma_f32<<<64, 32, 0, stream>>>(We1, H, Wa0 + 2*H*H,  H, Ca,       H,   H, H, 0, nullptr, nullptr, nullptr);
        gemm_wmma_f32<<<64, 32, 0, stream>>>(We1, H, Wm0 + H * H,  H, Cm,       H,   H, H, 0, nullptr, nullptr, nullptr);
        fold_bias_kernel<<<1, H, 0, stream>>>(bq, bk, bv, be1, Wa0, ba0, Wm0, bm0, Battn, Bmsg);

        // --- node projections: PN[N,384] = h @ [Wqa|Wka|Wvm] ---
        gemm_wmma_f32<<<tilesM_nodes * 24, 32, 0, stream>>>(
            hin, H, W3, 384, PNbuf, 384, NN, 384, 0, nullptr, nullptr, nullptr);

        init_accum_kernel<<<(NN * H + 255) / 256, 256, 0, stream>>>(Sbuf, wsum, den, maxint, NN);

        // --- pass 1: attention scores + segment max ---
        for (int cs = 0; cs < E; cs += CHUNK) {
            int nE = E - cs; if (nE > CHUNK) nE = CHUNK;
            int tm = (nE + 15) / 16;
            edge_gemm_wmma<<<tm, 256, 0, stream>>>(ea, cs, nE, w0, b0, Ca, Tc);
            attn_s_kernel<<<nE, 32, 0, stream>>>(PNbuf, Tc, ei, E, cs, Battn, Wa1, ba1, sbuf, maxint);
        }
        expden_kernel<<<(E + 255) / 256, 256, 0, stream>>>(sbuf, ei, E, maxint, den, exbuf);

        // --- pass 2: weighted message scatter ---
        for (int cs = 0; cs < E; cs += CHUNK) {
            int nE = E - cs; if (nE > CHUNK) nE = CHUNK;
            int tm = (nE + 15) / 16;
            edge_gemm_wmma<<<tm, 256, 0, stream>>>(ea, cs, nE, w0, b0, Cm, Tc);
            msg_scatter_kernel<<<nE, 32, 0, stream>>>(PNbuf, Tc, ei, E, cs, Bmsg, exbuf, den, Sbuf, wsum);
        }

        // --- node update: h_out = S@Wm1 + h_in + wsum*bm1 ---
        gemm_wmma_f32<<<tilesM_nodes * 8, 32, 0, stream>>>(
            Sbuf, H, Wm1, H, hout, H, NN, H, 1, hin, wsum, bm1);

        float* t = hin; hin = hout; hout = t;
    }

    // --- mean pool per graph ---
    zero_pool_kernel<<<(G * H + 255) / 256, 256, 0, stream>>>(hs, cnt, G);
    pool_scatter_kernel<<<NN, 32, 0, stream>>>(hin, batch, hs, cnt);
    pool_final_kernel<<<(G * H + 255) / 256, 256, 0, stream>>>(hs, cnt, hg, G);

    // --- heads (leaf order: Aeff, Disp, GVD, NL, neff; output cols 1,3,4,2,0) ---
    const int hbase[5] = { 92, 100, 106, 112, 120 };
    const int hnl[5]   = { 4, 3, 3, 4, 3 };
    const int hcol[5]  = { 1, 3, 4, 2, 0 };
    for (int hd = 0; hd < 5; ++hd) {
        HeadArgs a = {};
        int nb = hbase[hd];
        a.W0 = P(nb + 0); a.B0 = P(nb + 1);
        a.W1 = P(nb + 2); a.B1 = P(nb + 3);
        a.W2 = P(nb + 4); a.B2 = P(nb + 5);
        if (hnl[hd] == 4) { a.W3 = P(nb + 6); a.B3 = P(nb + 7); }
        else              { a.W3 = P(nb + 4); a.B3 = P(nb + 5); }
        a.nl = hnl[hd]; a.col = hcol[hd];
        head_kernel<<<G, H, 0, stream>>>(hg, a, (float*)d_out);
    }
}